// NNUE_17549236372205
// MI455X (gfx1250) — compile-verified
//
#include <hip/hip_runtime.h>

// ---------------------------------------------------------------------------
// NNUE forward pass for MI455X (gfx1250, wave32).
// Layer 0 (memory-bound, ~671 MB streamed, ~29us floor at 23.3 TB/s) uses
// V_WMMA_F32_16X16X4_F32:
//   A  = W0 tile  [16(M, rows 0..3 real) x 4(K)]   -> v2f per lane
//   B  = features [4(K) x 16(N = batch rows)]      -> v2f per lane
//   D += A x B    [16 x 16], rows 0..3 are the 4 layer-0 outputs.
// K split 32 ways across waves; partials combined via global_atomic_add_f32.
// Feature stream is non-temporal (read-once); W0 stays L2-resident.
// ---------------------------------------------------------------------------

typedef __attribute__((ext_vector_type(2))) float v2f;
typedef __attribute__((ext_vector_type(4))) float v4f;
typedef __attribute__((ext_vector_type(8))) float v8f;

constexpr int kNF     = 40960;            // NUM_FEATURES
constexpr int kB      = 2048;             // batch
constexpr int kM      = 4;                // layer-0 outputs
constexpr int kKSplit = 32;               // K-chunks per row tile
constexpr int kKC     = kNF / kKSplit;    // 1280 floats per chunk (40 x 32)
constexpr int kTiles  = (2 * kB) / 16;    // 256 row tiles (white 0..2047, black 2048..4095)

__global__ void zero_ws_kernel(float* __restrict__ p, int n) {
  int i = blockIdx.x * blockDim.x + threadIdx.x;
  if (i < n) p[i] = 0.0f;
}

__global__ __launch_bounds__(256)
void nnue_l0_wmma(const float* __restrict__ wf,
                  const float* __restrict__ bf,
                  const float* __restrict__ W0,
                  float* __restrict__ accbuf) {
  const int wave = (blockIdx.x * blockDim.x + threadIdx.x) >> 5;
  const int lane = threadIdx.x & 31;
  // Force the per-wave scalars into SGPRs so loop control is SALU, not VALU.
  const int tile  = __builtin_amdgcn_readfirstlane(wave / kKSplit);   // 0..255
  const int chunk = __builtin_amdgcn_readfirstlane(wave % kKSplit);   // 0..31
  const int hi    = lane >> 4;                // half-wave: K-offset selector
  const int idx   = lane & 15;                // B: batch row in tile; A: output m

  // Feature row for the B fragment (every lane streams its own row).
  const int grow = tile * 16 + idx;           // global row 0..4095
  const float* feat = (grow < kB) ? (wf + (size_t)grow * kNF)
                                  : (bf + (size_t)(grow - kB) * kNF);

  // Weight row for the A fragment (only m < 4 is real; others masked to 0).
  const bool  wvalid = (idx < kM);
  const float wmask  = wvalid ? 1.0f : 0.0f;
  const float* wrow  = W0 + (size_t)(wvalid ? idx : 0) * kNF;

  const int k0   = __builtin_amdgcn_readfirstlane(chunk * kKC);
  const int kend = k0 + kKC;

  v8f acc = {};

  // Outer trip covers 32 K values: 1 speculative prefetch (~4KB ahead; past-
  // end prefetches are dropped per ISA 10.5), 8 b128 loads, 8 WMMAs.
  // K-slot remap (reduction is order independent):
  //   WMMA#1 slots {2hi,2hi+1} <-> memory {k+4hi,   k+4hi+1} (.xy of the b128)
  //   WMMA#2 slots {2hi,2hi+1} <-> memory {k+4hi+2, k+4hi+3} (.zw of the b128)
  // A and B use the identical mapping, so the dot products line up.
  for (int k = k0; k < kend; k += 32) {
    __builtin_prefetch(feat + k + 1024, 0, 1);   // global_prefetch_b8
#pragma unroll
    for (int u = 0; u < 32; u += 8) {
      const int koff = k + u + hi * 4;
      v4f f = __builtin_nontemporal_load((const v4f*)(feat + koff));  // NT stream
      v4f w = *(const v4f*)(wrow + koff);   // masked lanes read row 0 (L2 hit)
      w *= wmask;                           // zero-pad M rows 4..15 branch-free

      v2f a0 = {w.x, w.y}, f0 = {f.x, f.y};
      v2f a1 = {w.z, w.w}, f1 = {f.z, f.w};
      acc = __builtin_amdgcn_wmma_f32_16x16x4_f32(false, a0, false, f0,
                                                  (short)0, acc, false, false);
      acc = __builtin_amdgcn_wmma_f32_16x16x4_f32(false, a1, false, f1,
                                                  (short)0, acc, false, false);
    }
  }

  // D layout: lanes 0..15, VGPR v holds M=v, N=lane -> acc[0..3] are the
  // 4 layer-0 partial outputs for batch row `grow`.
  if (hi == 0) {
    float* dst = accbuf + (size_t)grow * kM;
    atomicAdd(dst + 0, acc[0]);
    atomicAdd(dst + 1, acc[1]);
    atomicAdd(dst + 2, acc[2]);
    atomicAdd(dst + 3, acc[3]);
  }
}

__global__ __launch_bounds__(256)
void nnue_tail(const float* __restrict__ accbuf,
               const int*   __restrict__ turn,
               const float* __restrict__ b0v,
               const float* __restrict__ W1, const float* __restrict__ b1v,
               const float* __restrict__ W2, const float* __restrict__ b2v,
               float* __restrict__ out) {
  const int i = blockIdx.x * blockDim.x + threadIdx.x;
  if (i >= kB) return;

  float wv[4], bv[4];
#pragma unroll
  for (int j = 0; j < 4; ++j) {
    wv[j] = accbuf[(size_t)i * 4 + j]        + b0v[j];
    bv[j] = accbuf[(size_t)(kB + i) * 4 + j] + b0v[j];
  }

  const float t = (float)turn[i];
  float a[8];
#pragma unroll
  for (int j = 0; j < 4; ++j) {
    a[j]     = t * wv[j] + (1.0f - t) * bv[j];
    a[4 + j] = t * bv[j] + (1.0f - t) * wv[j];
  }
#pragma unroll
  for (int j = 0; j < 8; ++j) a[j] = fminf(fmaxf(a[j], 0.0f), 1.0f);

  float r = b2v[0];
#pragma unroll
  for (int o = 0; o < 8; ++o) {
    float s = b1v[o];
#pragma unroll
    for (int j = 0; j < 8; ++j) s = fmaf(W1[o * 8 + j], a[j], s);
    s = fminf(fmaxf(s, 0.0f), 1.0f);
    r = fmaf(W2[o], s, r);
  }
  out[i] = r;
}

extern "C" void kernel_launch(void* const* d_in, const int* in_sizes, int n_in,
                              void* d_out, int out_size, void* d_ws, size_t ws_size,
                              hipStream_t stream) {
  (void)in_sizes; (void)n_in; (void)out_size; (void)ws_size;
  const float* wf   = (const float*)d_in[0];   // white_features [B, NF]
  const float* bf   = (const float*)d_in[1];   // black_features [B, NF]
  const int*   turn = (const int*)  d_in[2];   // [B, 1]
  // d_in[3] score, d_in[4] result: unused by the forward pass
  const float* W0 = (const float*)d_in[5];     // [4, NF]
  const float* b0 = (const float*)d_in[6];     // [4]
  const float* W1 = (const float*)d_in[7];     // [8, 8]
  const float* b1 = (const float*)d_in[8];     // [8]
  const float* W2 = (const float*)d_in[9];     // [1, 8]
  const float* b2 = (const float*)d_in[10];    // [1]

  float* accbuf = (float*)d_ws;                // 2*B*4 floats = 64 KB of scratch
  const int nacc = 2 * kB * kM;

  zero_ws_kernel<<<(nacc + 255) / 256, 256, 0, stream>>>(accbuf, nacc);

  const int waves  = kTiles * kKSplit;         // 8192 waves (~8 per SIMD32)
  const int blocks = waves / 8;                // 256 threads = 8 wave32s per block
  nnue_l0_wmma<<<blocks, 256, 0, stream>>>(wf, bf, W0, accbuf);

  nnue_tail<<<(kB + 255) / 256, 256, 0, stream>>>(accbuf, turn, b0, W1, b1, W2, b2,
                                                  (float*)d_out);
}